// FastGRNNPredictor_13331578487077
// MI455X (gfx1250) — compile-verified
//
#include <hip/hip_runtime.h>
#include <math.h>

#define HID   100
#define CHUNK 32
#define STEPS 99
#define WRANK 16
#define URANK 32
#define NCLS  10
#define HIDP  112   // HID padded to multiple of 16 for guard-free WMMA N-tiles
#define HPAD  128   // hidden dim padded to 4*wave32 for guard-free scan lanes
#define ROWSP 112   // STEPS padded to multiple of 16 for guard-free stores

typedef __attribute__((ext_vector_type(2))) float v2f;
typedef __attribute__((ext_vector_type(8))) float v8f;

__device__ __forceinline__ float sigmoidf_fast(float v) {
    return 1.0f / (1.0f + __expf(-v));
}

// ---------------------------------------------------------------------------
// Kernel 1: A[112,128-stride] = X[99,32] @ W1[32,16] @ W2[16,100] via f32
// WMMA. One wave per 16-row tile. Operands staged in LDS (guards applied at
// staging only), B-operands transposed [n][k] so every WMMA operand fetch is
// one aligned ds_load_b64. Output stored PADDED so all stores are
// unconditional — no divergent branches anywhere after staging.
//
// V_WMMA_F32_16X16X4_F32 layouts (wave32):
//   A 16x4 : lane L -> row M=L%16; VGPR0/1 hold K pair; upper half-wave K+2
//   B 4x16 : lane L -> col N=L%16; same K striping
//   C/D    : 8 VGPRs; VGPR i -> row i (lanes 0-15) / row i+8 (lanes 16-31)
// ---------------------------------------------------------------------------
__global__ __launch_bounds__(32)
void fastgrnn_precompute_wmma(const float* __restrict__ x,
                              const float* __restrict__ W1,
                              const float* __restrict__ W2,
                              float* __restrict__ A)
{
    __shared__ float Xs [16 * CHUNK];     // 2.0 KB, row-major [m][k]
    __shared__ float W1t[WRANK * CHUNK];  // 2.0 KB, transposed [n][k]
    __shared__ float W2t[HIDP * WRANK];   // 7.0 KB, transposed [col][k], zero-pad
    __shared__ float Tl [16 * WRANK];     // 1.0 KB, row-major [m][k]

    const int lane  = threadIdx.x;        // 0..31
    const int mn    = lane & 15;          // row (A-op) / col (B-op) in tile
    const int khalf = (lane >> 4) << 1;   // 0 or 2
    const int m0    = blockIdx.x * 16;    // tile row base (step index)

    // ---- stage operands into LDS (guards live here, once) ----
    for (int i = lane; i < 16 * CHUNK; i += 32) {
        const int r = m0 + (i >> 5);
        Xs[i] = (r < STEPS) ? x[r * CHUNK + (i & 31)] : 0.0f;
    }
    for (int i = lane; i < WRANK * CHUNK; i += 32) {       // W1 is [k][n]
        const int k = i >> 4, n = i & 15;
        W1t[n * CHUNK + k] = W1[i];
    }
    for (int i = lane; i < HIDP * WRANK; i += 32) {        // W2 is [k][col]
        const int c = i >> 4, k = i & 15;
        W2t[i] = (c < HID) ? W2[k * HID + c] : 0.0f;
    }
    __syncthreads();

    // ---- stage 1: T[16,16] = Xtile[16,32] @ W1[32,16], K=32 in 8 wmma steps
    v8f acc = {};
    #pragma unroll
    for (int kk = 0; kk < CHUNK; kk += 4) {
        const int k = kk + khalf;
        v2f a = *(const v2f*)&Xs [mn * CHUNK + k];
        v2f b = *(const v2f*)&W1t[mn * CHUNK + k];
        acc = __builtin_amdgcn_wmma_f32_16x16x4_f32(false, a, false, b,
                                                    (short)0, acc, false, false);
    }

    // D-layout -> row-major [m][k] in LDS for stage-2 A-operand reads
    #pragma unroll
    for (int i = 0; i < 8; ++i) {
        const int r = i + ((lane >= 16) ? 8 : 0);
        Tl[r * WRANK + mn] = acc[i];
    }
    __syncthreads();

    // ---- stage 2: Atile[16,112] = T[16,16] @ W2pad[16,112], 7 N-tiles ----
    for (int n0 = 0; n0 < HIDP; n0 += 16) {
        const int col = n0 + mn;
        v8f acc2 = {};
        #pragma unroll
        for (int kk = 0; kk < WRANK; kk += 4) {
            const int k = kk + khalf;
            v2f a = *(const v2f*)&Tl [mn  * WRANK + k];
            v2f b = *(const v2f*)&W2t[col * WRANK + k];
            acc2 = __builtin_amdgcn_wmma_f32_16x16x4_f32(false, a, false, b,
                                                         (short)0, acc2, false, false);
        }
        // unconditional stores into the padded A (stride HPAD)
        #pragma unroll
        for (int i = 0; i < 8; ++i) {
            const int r = m0 + i + ((lane >= 16) ? 8 : 0);
            A[r * HPAD + col] = acc2[i];
        }
    }

    // zero the tail columns 112..127 of this tile's rows (deterministic pad)
    for (int i = lane; i < 16 * (HPAD - HIDP); i += 32) {
        const int r = m0 + (i >> 4);
        A[r * HPAD + HIDP + (i & 15)] = 0.0f;
    }
}

// ---------------------------------------------------------------------------
// Kernel 2: the sequential 99-step scan. Single wave; all recurrent weights
// in LDS, hidden dim padded to 128 so every lane owns exactly 4 columns —
// no divergent trip counts in the step loop. Pad columns compute dead values
// that never feed the live 100 dims (the H@U1 reduction reads i<100 only).
// A[t+1] is prefetched one step ahead (global_prefetch_b8).
// ---------------------------------------------------------------------------
__global__ __launch_bounds__(32)
void fastgrnn_recurrence(const float* __restrict__ A,
                         const float* __restrict__ U1,
                         const float* __restrict__ U2,
                         const float* __restrict__ bg,
                         const float* __restrict__ bu,
                         const float* __restrict__ zeta,
                         const float* __restrict__ nu,
                         const float* __restrict__ FC,
                         const float* __restrict__ FCb,
                         float* __restrict__ out, int out_size)
{
    __shared__ float U1s[HID * URANK];    // 12.8 KB, [i][j]
    __shared__ float U2p[URANK * HPAD];   // 16.0 KB, [j][k], zero-padded cols
    __shared__ float FCs[HID * NCLS];     //  4.0 KB
    __shared__ float Hs [HPAD];
    __shared__ float Ts [URANK];
    __shared__ float bgs[HPAD];
    __shared__ float bus[HPAD];
    __shared__ float score_s[NCLS];

    const int lane = threadIdx.x;

    for (int i = lane; i < HID * URANK; i += 32) U1s[i] = U1[i];
    for (int i = lane; i < URANK * HPAD; i += 32) {
        const int j = i >> 7, k = i & (HPAD - 1);
        U2p[i] = (k < HID) ? U2[j * HID + k] : 0.0f;
    }
    for (int i = lane; i < HID * NCLS; i += 32) FCs[i] = FC[i];
    for (int i = lane; i < HPAD; i += 32) {
        bgs[i] = (i < HID) ? bg[i] : 0.0f;
        bus[i] = (i < HID) ? bu[i] : 0.0f;
        Hs[i]  = 0.0f;
    }
    __syncthreads();

    const float sz = sigmoidf_fast(zeta[0]);
    const float sn = sigmoidf_fast(nu[0]);

    for (int t = 0; t < STEPS; ++t) {
        // pull next step's A row (512 B) toward the WGP while computing
        if ((t + 1) < STEPS && lane < 16)
            __builtin_prefetch(&A[(t + 1) * HPAD] + lane * 8, 0, 0);

        // t_j = (H @ U1)[j], one output per lane (column reads, no conflicts)
        float tj = 0.0f;
        #pragma unroll 4
        for (int i = 0; i < HID; ++i)
            tj = fmaf(Hs[i], U1s[i * URANK + lane], tj);
        Ts[lane] = tj;
        __syncthreads();

        // each lane owns columns lane, lane+32, lane+64, lane+96 (uniform)
        float hn[4];
        #pragma unroll
        for (int q = 0; q < 4; ++q) {
            const int k = lane + 32 * q;
            float b = 0.0f;
            #pragma unroll
            for (int j = 0; j < URANK; ++j)
                b = fmaf(Ts[j], U2p[j * HPAD + k], b);
            const float c = A[t * HPAD + k] + b;
            const float g = sigmoidf_fast(c + bgs[k]);
            const float h = tanhf(c + bus[k]);
            hn[q] = g * Hs[k] + (sz * (1.0f - g) + sn) * h;
        }
        __syncthreads();                 // everyone done reading old H
        #pragma unroll
        for (int q = 0; q < 4; ++q) Hs[lane + 32 * q] = hn[q];
        __syncthreads();                 // new H visible
    }

    // score = H @ FC + FCbias ; lanes 0..9 own one class each
    if (lane < NCLS) {
        float s = FCb[lane];
        #pragma unroll 4
        for (int i = 0; i < HID; ++i)
            s = fmaf(Hs[i], FCs[i * NCLS + lane], s);
        score_s[lane] = s;
        if (lane < out_size) out[lane] = s;
    }
    __syncthreads();
    if (lane == 0 && out_size > NCLS) {
        int   am   = 0;
        float best = score_s[0];
        for (int k = 1; k < NCLS; ++k)
            if (score_s[k] > best) { best = score_s[k]; am = k; }
        out[NCLS] = (float)am;           // pred, flattened after score
    }
}

extern "C" void kernel_launch(void* const* d_in, const int* in_sizes, int n_in,
                              void* d_out, int out_size, void* d_ws, size_t ws_size,
                              hipStream_t stream) {
    (void)in_sizes; (void)n_in; (void)ws_size;
    const float* x    = (const float*)d_in[0];
    const float* W1   = (const float*)d_in[1];
    const float* W2   = (const float*)d_in[2];
    const float* U1   = (const float*)d_in[3];
    const float* U2   = (const float*)d_in[4];
    const float* bg   = (const float*)d_in[5];
    const float* bu   = (const float*)d_in[6];
    const float* zeta = (const float*)d_in[7];
    const float* nu   = (const float*)d_in[8];
    const float* FC   = (const float*)d_in[9];
    const float* FCb  = (const float*)d_in[10];

    float* A   = (float*)d_ws;           // padded A[112][128] scratch, 56 KB
    float* out = (float*)d_out;

    dim3 grid1(ROWSP / 16);              // 7 tiles of 16 steps
    fastgrnn_precompute_wmma<<<grid1, 32, 0, stream>>>(x, W1, W2, A);
    fastgrnn_recurrence<<<1, 32, 0, stream>>>(A, U1, U2, bg, bu, zeta, nu,
                                              FC, FCb, out, out_size);
}